// OCGraphSAGE_51616916963801
// MI455X (gfx1250) — compile-verified
//
#include <hip/hip_runtime.h>
#include <hip/hip_bf16.h>

typedef __attribute__((ext_vector_type(2))) float v2f;
typedef __attribute__((ext_vector_type(8))) float v8f;

#define IN_CH 64
#define WPB   4   // waves per block in the WMMA kernels

// ---------------------------------------------------------------- utilities
__global__ void zero_f32_kernel(float* __restrict__ p, long long n) {
    long long i = (long long)blockIdx.x * blockDim.x + threadIdx.x;
    if (i < n) p[i] = 0.0f;
}

__global__ void degree_kernel(const int* __restrict__ dst, float* __restrict__ deg,
                              long long E) {
    long long e = (long long)blockIdx.x * blockDim.x + threadIdx.x;
    if (e < E) atomicAdd(&deg[dst[e]], 1.0f);
}

// 16 threads per edge, 4 channels each (float4 gather, 4 f32 atomics)
__global__ void scatter_add_kernel(const float* __restrict__ feat,
                                   const int* __restrict__ src,
                                   const int* __restrict__ dst,
                                   float* __restrict__ agg, long long E) {
    long long gid = (long long)blockIdx.x * blockDim.x + threadIdx.x;
    if (gid >= E * 16) return;
    int e = (int)(gid >> 4);
    int c = (int)(gid & 15) * 4;
    int s = src[e], d = dst[e];
    const float4 v = *(const float4*)(feat + (size_t)s * IN_CH + c);
    float* o = agg + (size_t)d * IN_CH + c;
    atomicAdd(o + 0, v.x);
    atomicAdd(o + 1, v.y);
    atomicAdd(o + 2, v.z);
    atomicAdd(o + 3, v.w);
}

// ------------------------------------------------- fused SAGE linear (WMMA)
// out[n,:] = act( mean[n,:] @ wl^T + hin[n,:] @ wr^T + bias ),  K = 64
// mean[n,k] = agg[n,k] / max(deg[n],1)   (only when DUAL)
template <int NOUT, bool DUAL, bool RELU>
__global__ void sage_linear_kernel(const float* __restrict__ hin,
                                   const float* __restrict__ agg,
                                   const float* __restrict__ deg,
                                   const float* __restrict__ wl,   // [NOUT][64]
                                   const float* __restrict__ wr,   // [NOUT][64]
                                   const float* __restrict__ bias, // [NOUT]
                                   float* __restrict__ out,        // [N][NOUT]
                                   int nNodes) {
    constexpr int NT = NOUT / 16;
    __shared__ float wl_s[NOUT][66];                 // stride 66: 8B-aligned pairs,
    __shared__ float wr_s[(DUAL ? NOUT : 1)][66];    // bank-conflict free
    __shared__ float b_s[NOUT];

    // stage weights + bias into LDS
    for (int i = threadIdx.x; i < NOUT * IN_CH; i += blockDim.x) {
        wl_s[i >> 6][i & 63] = wl[i];
        if (DUAL) wr_s[i >> 6][i & 63] = wr[i];
    }
    for (int i = threadIdx.x; i < NOUT; i += blockDim.x) b_s[i] = bias[i];
    __syncthreads();

    const int wave = threadIdx.x >> 5;
    const int lane = threadIdx.x & 31;
    const int tile = blockIdx.x * WPB + wave;
    const int ntiles = (nNodes + 15) >> 4;
    if (tile >= ntiles) return;          // wave-uniform: EXEC all-ones below

    const int m  = lane & 15;            // A row / B col / D col
    const int hi = lane >> 4;            // half-wave select
    const int row  = tile * 16 + m;
    const int rrow = row < nNodes ? row : nNodes - 1;

    const float* hRow = hin + (size_t)rrow * IN_CH;
    const float* aRow = DUAL ? (agg + (size_t)rrow * IN_CH) : hRow;
    const float scale = DUAL ? (1.0f / fmaxf(deg[rrow], 1.0f)) : 1.0f;

    v8f acc[NT] = {};

#pragma unroll
    for (int kk = 0; kk < IN_CH / 4; ++kk) {
        const int k = kk * 4 + hi * 2;   // lane<16 -> K={0,1}; lane>=16 -> K={2,3}
        v2f ax;                          // root-feature A fragment
        {
            float2 t = *(const float2*)(hRow + k);
            ax.x = t.x; ax.y = t.y;
        }
        v2f am;                          // mean-aggregate A fragment
        if (DUAL) {
            float2 u = *(const float2*)(aRow + k);
            am.x = u.x * scale; am.y = u.y * scale;
        }
#pragma unroll
        for (int nt = 0; nt < NT; ++nt) {
            const int j = nt * 16 + m;   // output channel handled by this lane
            if (DUAL) {
                v2f bl;
                {
                    float2 wv = *(const float2*)&wl_s[j][k];
                    bl.x = wv.x; bl.y = wv.y;
                }
                acc[nt] = __builtin_amdgcn_wmma_f32_16x16x4_f32(
                    false, am, false, bl, (short)0, acc[nt], false, false);
                v2f br;
                {
                    float2 wv = *(const float2*)&wr_s[j][k];
                    br.x = wv.x; br.y = wv.y;
                }
                acc[nt] = __builtin_amdgcn_wmma_f32_16x16x4_f32(
                    false, ax, false, br, (short)0, acc[nt], false, false);
            } else {
                v2f bl;
                {
                    float2 wv = *(const float2*)&wl_s[j][k];
                    bl.x = wv.x; bl.y = wv.y;
                }
                acc[nt] = __builtin_amdgcn_wmma_f32_16x16x4_f32(
                    false, ax, false, bl, (short)0, acc[nt], false, false);
            }
        }
    }

    // epilogue: D layout -> VGPR r holds M = hi*8 + r, N = lane&15
#pragma unroll
    for (int nt = 0; nt < NT; ++nt) {
#pragma unroll
        for (int r = 0; r < 8; ++r) {
            const int mm = hi * 8 + r;
            const int orow = tile * 16 + mm;
            if (orow < nNodes) {
                float v = acc[nt][r] + b_s[nt * 16 + m];
                if (RELU) v = fmaxf(v, 0.0f);
                out[(size_t)orow * NOUT + nt * 16 + m] = v;
            }
        }
    }
}

// ------------------------------------------------------------------ launch
extern "C" void kernel_launch(void* const* d_in, const int* in_sizes, int n_in,
                              void* d_out, int out_size, void* d_ws, size_t ws_size,
                              hipStream_t stream) {
    const float* x   = (const float*)d_in[0];
    const int*   ei  = (const int*)d_in[1];
    const float* w1l = (const float*)d_in[2];
    const float* b1  = (const float*)d_in[3];
    const float* w1r = (const float*)d_in[4];
    const float* w2l = (const float*)d_in[5];
    const float* b2  = (const float*)d_in[6];
    const float* w2r = (const float*)d_in[7];
    const float* wh  = (const float*)d_in[8];
    const float* bh  = (const float*)d_in[9];

    const int       N = in_sizes[0] / IN_CH;       // 100000
    const long long E = (long long)in_sizes[1] / 2; // 1600000
    const int* src = ei;
    const int* dst = ei + E;

    // workspace layout
    float* agg = (float*)d_ws;                        // N*64
    float* h1  = agg + (size_t)N * IN_CH;             // N*64
    float* h2  = h1 + (size_t)N * IN_CH;              // N*64
    float* deg = h2 + (size_t)N * IN_CH;              // N

    const long long nfeat = (long long)N * IN_CH;
    const int ZB = 256;
    const int ntiles = (N + 15) / 16;
    const int gBlocks = (ntiles + WPB - 1) / WPB;
    const long long sThreads = E * 16;

    // ---- degree (once, shared by both layers) ----
    zero_f32_kernel<<<(int)((N + ZB - 1) / ZB), ZB, 0, stream>>>(deg, N);
    degree_kernel<<<(int)((E + ZB - 1) / ZB), ZB, 0, stream>>>(dst, deg, E);

    // ---- layer 1 ----
    zero_f32_kernel<<<(int)((nfeat + ZB - 1) / ZB), ZB, 0, stream>>>(agg, nfeat);
    scatter_add_kernel<<<(int)((sThreads + ZB - 1) / ZB), ZB, 0, stream>>>(x, src, dst, agg, E);
    sage_linear_kernel<64, true, true><<<gBlocks, WPB * 32, 0, stream>>>(
        x, agg, deg, w1l, w1r, b1, h1, N);

    // ---- layer 2 ----
    zero_f32_kernel<<<(int)((nfeat + ZB - 1) / ZB), ZB, 0, stream>>>(agg, nfeat);
    scatter_add_kernel<<<(int)((sThreads + ZB - 1) / ZB), ZB, 0, stream>>>(h1, src, dst, agg, E);
    sage_linear_kernel<64, true, true><<<gBlocks, WPB * 32, 0, stream>>>(
        h1, agg, deg, w2l, w2r, b2, h2, N);

    // ---- head: out = h2 @ wh^T + bh  (no relu) ----
    sage_linear_kernel<32, false, false><<<gBlocks, WPB * 32, 0, stream>>>(
        h2, h2, deg, wh, wh, bh, (float*)d_out, N);
}